// MoEEncoderDecoderGPT_64089501991423
// MI455X (gfx1250) — compile-verified
//
#include <hip/hip_runtime.h>
#include <math.h>

// ---------------- problem constants (match reference) ----------------
constexpr int Bb = 4, Ss = 2048, Dd = 1024;
constexpr int Hh = 2 * Dd;           // 2048
constexpr int Aa = Hh / 16;          // 128
constexpr int Ee = 8;
constexpr int Tt = Bb * Ss;          // 8192

typedef __attribute__((ext_vector_type(2))) float v2f;
typedef __attribute__((ext_vector_type(4))) float v4f;
typedef __attribute__((ext_vector_type(8))) float v8f;
typedef __attribute__((ext_vector_type(4))) int v4i;

#define DEVI static __device__ __forceinline__

// ---- CDNA5 async global->LDS (ASYNCcnt-tracked), guarded by __has_builtin
#if defined(__has_builtin)
#if __has_builtin(__builtin_amdgcn_global_load_async_to_lds_b128)
#define HAVE_ASYNC_LDS 1
#endif
#if __has_builtin(__builtin_amdgcn_s_wait_asynccnt)
#define HAVE_WAIT_ASYNC 1
#endif
#endif

#ifdef HAVE_ASYNC_LDS
// builtin signature (from probe): (v4i AS1*, v4i AS3*, imm int offset, imm cpol)
typedef __attribute__((address_space(1))) v4i* gp4i;
typedef __attribute__((address_space(3))) v4i* lp4i;
DEVI void async_g2l_b128(const float* g, float* l) {
  __builtin_amdgcn_global_load_async_to_lds_b128(
      (gp4i)(uintptr_t)g, (lp4i)(uint32_t)(uintptr_t)l, 0, 0);
}
#endif
DEVI void wait_async0() {
#ifdef HAVE_WAIT_ASYNC
  __builtin_amdgcn_s_wait_asynccnt(0);
#else
  asm volatile("s_wait_asynccnt 0x0" ::: "memory");
#endif
}
DEVI void wait_async6() {
#ifdef HAVE_WAIT_ASYNC
  __builtin_amdgcn_s_wait_asynccnt(6);
#else
  asm volatile("s_wait_asynccnt 0x6" ::: "memory");
#endif
}

// FP32 WMMA: D(16x16,f32) = A(16x4,f32) * B(4x16,f32) + C
DEVI v8f wmma4(v2f a, v2f b, v8f c) {
  return __builtin_amdgcn_wmma_f32_16x16x4_f32(false, a, false, b, (short)0, c,
                                               false, false);
}
DEVI float siluf(float x) { return x / (1.0f + expf(-x)); }

DEVI float wave_sum32(float v) {
#pragma unroll
  for (int m = 16; m > 0; m >>= 1) v += __shfl_xor(v, m, 32);
  return v;
}
// reduce across the 16 lanes of each half-wave (masks 1,2,4,8)
DEVI float half_sum16(float v) {
#pragma unroll
  for (int m = 8; m > 0; m >>= 1) v += __shfl_xor(v, m, 32);
  return v;
}

// =====================================================================
// Generic C[M,N] = A[M,K] @ W[N,K]^T  (row-major everything)
// block: 256 thr = 8 waves; block tile 128x64; wave tile 32x32 (2x2 WMMA)
// Double-buffered LDS; async global->LDS copies when available, else
// register-staged pipelining (load next chunk while computing current).
// =====================================================================
enum { EP_STORE = 0, EP_ADDSC = 1, EP_FINAL = 2 };
// EP_STORE: C = acc + bias[n]
// EP_ADDSC: C += alpha*acc
// EP_FINAL: C = C*rowscale[m] + alpha*acc      (in-place on C)

__global__ __launch_bounds__(256) void gemm_xwt(
    const float* __restrict__ A, const float* __restrict__ W,
    const float* __restrict__ bias, float* __restrict__ C,
    const float* __restrict__ rowscale, int M, int N, int K, float alpha,
    int mode) {
#ifdef HAVE_ASYNC_LDS
  __shared__ float As[2][128 * 36];  // stride 36: conflict-free, 16B rows
  __shared__ float Ws[2][64 * 36];
#else
  __shared__ float As1[128 * 36];
  __shared__ float Ws1[64 * 36];
#endif
  const int tid = threadIdx.x;
  const int lane = tid & 31, wave = tid >> 5;
  const int lm = lane & 15, half = lane >> 4;
  const int wm = wave >> 1, wn = wave & 1;
  const int m0 = blockIdx.y * 128, n0 = blockIdx.x * 64;

  v8f acc[2][2] = {};

#ifdef HAVE_ASYNC_LDS
  // ---------------- async double-buffered pipeline ----------------
  auto stage = [&](int buf, int k0) {
#pragma unroll
    for (int i = 0; i < 4; ++i) {  // A tile 128x32: 1024 float4 / 256 thr
      int idx = tid + 256 * i;
      int r = idx >> 3, c4 = (idx & 7) << 2;
      async_g2l_b128(&A[(size_t)(m0 + r) * K + k0 + c4], &As[buf][r * 36 + c4]);
    }
#pragma unroll
    for (int i = 0; i < 2; ++i) {  // W tile 64x32
      int idx = tid + 256 * i;
      int r = idx >> 3, c4 = (idx & 7) << 2;
      async_g2l_b128(&W[(size_t)(n0 + r) * K + k0 + c4], &Ws[buf][r * 36 + c4]);
    }
  };
  stage(0, 0);
  for (int k0 = 0; k0 < K; k0 += 32) {
    const int buf = (k0 >> 5) & 1;
    if (k0 + 32 < K) {
      stage(buf ^ 1, k0 + 32);  // prefetch next chunk (6 more async ops)
      wait_async6();            // previous 6 (current buf) retired
    } else {
      wait_async0();
    }
    __syncthreads();
#pragma unroll
    for (int kk = 0; kk < 32; kk += 4) {
      v2f a[2], b[2];
#pragma unroll
      for (int im = 0; im < 2; ++im)
        a[im] =
            *(const v2f*)&As[buf][(32 * wm + 16 * im + lm) * 36 + kk + 2 * half];
#pragma unroll
      for (int in = 0; in < 2; ++in)
        b[in] =
            *(const v2f*)&Ws[buf][(32 * wn + 16 * in + lm) * 36 + kk + 2 * half];
#pragma unroll
      for (int im = 0; im < 2; ++im)
#pragma unroll
        for (int in = 0; in < 2; ++in)
          acc[im][in] = wmma4(a[im], b[in], acc[im][in]);
    }
    __syncthreads();
  }
#else
  // ---------------- register-staged pipeline (fallback) ----------------
  v4f ta[4], tw[2];
  auto ldg = [&](int k0) {
#pragma unroll
    for (int i = 0; i < 4; ++i) {
      int idx = tid + 256 * i;
      int r = idx >> 3, c4 = (idx & 7) << 2;
      ta[i] = *(const v4f*)&A[(size_t)(m0 + r) * K + k0 + c4];
    }
#pragma unroll
    for (int i = 0; i < 2; ++i) {
      int idx = tid + 256 * i;
      int r = idx >> 3, c4 = (idx & 7) << 2;
      tw[i] = *(const v4f*)&W[(size_t)(n0 + r) * K + k0 + c4];
    }
  };
  ldg(0);
  for (int k0 = 0; k0 < K; k0 += 32) {
    __syncthreads();  // everyone done reading previous chunk
#pragma unroll
    for (int i = 0; i < 4; ++i) {
      int idx = tid + 256 * i;
      int r = idx >> 3, c4 = (idx & 7) << 2;
      *(v4f*)&As1[r * 36 + c4] = ta[i];
    }
#pragma unroll
    for (int i = 0; i < 2; ++i) {
      int idx = tid + 256 * i;
      int r = idx >> 3, c4 = (idx & 7) << 2;
      *(v4f*)&Ws1[r * 36 + c4] = tw[i];
    }
    __syncthreads();
    if (k0 + 32 < K) ldg(k0 + 32);  // overlap next loads with compute
#pragma unroll
    for (int kk = 0; kk < 32; kk += 4) {
      v2f a[2], b[2];
#pragma unroll
      for (int im = 0; im < 2; ++im)
        a[im] = *(const v2f*)&As1[(32 * wm + 16 * im + lm) * 36 + kk + 2 * half];
#pragma unroll
      for (int in = 0; in < 2; ++in)
        b[in] = *(const v2f*)&Ws1[(32 * wn + 16 * in + lm) * 36 + kk + 2 * half];
#pragma unroll
      for (int im = 0; im < 2; ++im)
#pragma unroll
        for (int in = 0; in < 2; ++in)
          acc[im][in] = wmma4(a[im], b[in], acc[im][in]);
    }
  }
#endif

#pragma unroll
  for (int im = 0; im < 2; ++im)
#pragma unroll
    for (int in = 0; in < 2; ++in)
#pragma unroll
      for (int i = 0; i < 8; ++i) {
        int row = m0 + 32 * wm + 16 * im + 8 * half + i;
        int col = n0 + 32 * wn + 16 * in + lm;
        size_t o = (size_t)row * N + col;
        float v = acc[im][in][i];
        if (mode == EP_STORE)
          C[o] = v + (bias ? bias[col] : 0.0f);
        else if (mode == EP_ADDSC)
          C[o] += alpha * v;
        else
          C[o] = C[o] * rowscale[row] + alpha * v;
      }
}

// =====================================================================
// Fused: hidden = silu(x@Wg^T + bg) * (x@Wu^T + bu)  (two GEMMs, one pass)
// single LDS buffer (36KB); async staging when available, else reg-staged
// =====================================================================
__global__ __launch_bounds__(256) void gemm_upgate(
    const float* __restrict__ X, const float* __restrict__ Wu,
    const float* __restrict__ bu, const float* __restrict__ Wgt,
    const float* __restrict__ bg, float* __restrict__ Hid, int M, int N,
    int K) {
  __shared__ float As[128 * 36];
  __shared__ float Us[64 * 36];
  __shared__ float Gs[64 * 36];
  const int tid = threadIdx.x;
  const int lane = tid & 31, wave = tid >> 5;
  const int lm = lane & 15, half = lane >> 4;
  const int wm = wave >> 1, wn = wave & 1;
  const int m0 = blockIdx.y * 128, n0 = blockIdx.x * 64;

  v8f au[2][2] = {}, ag[2][2] = {};
#ifndef HAVE_ASYNC_LDS
  v4f ta[4], tu[2], tg[2];
  auto ldg = [&](int k0) {
#pragma unroll
    for (int i = 0; i < 4; ++i) {
      int idx = tid + 256 * i;
      int r = idx >> 3, c4 = (idx & 7) << 2;
      ta[i] = *(const v4f*)&X[(size_t)(m0 + r) * K + k0 + c4];
    }
#pragma unroll
    for (int i = 0; i < 2; ++i) {
      int idx = tid + 256 * i;
      int r = idx >> 3, c4 = (idx & 7) << 2;
      tu[i] = *(const v4f*)&Wu[(size_t)(n0 + r) * K + k0 + c4];
      tg[i] = *(const v4f*)&Wgt[(size_t)(n0 + r) * K + k0 + c4];
    }
  };
  ldg(0);
#endif
  for (int k0 = 0; k0 < K; k0 += 32) {
    __syncthreads();
#ifdef HAVE_ASYNC_LDS
#pragma unroll
    for (int i = 0; i < 4; ++i) {
      int idx = tid + 256 * i;
      int r = idx >> 3, c4 = (idx & 7) << 2;
      async_g2l_b128(&X[(size_t)(m0 + r) * K + k0 + c4], &As[r * 36 + c4]);
    }
#pragma unroll
    for (int i = 0; i < 2; ++i) {
      int idx = tid + 256 * i;
      int r = idx >> 3, c4 = (idx & 7) << 2;
      async_g2l_b128(&Wu[(size_t)(n0 + r) * K + k0 + c4], &Us[r * 36 + c4]);
      async_g2l_b128(&Wgt[(size_t)(n0 + r) * K + k0 + c4], &Gs[r * 36 + c4]);
    }
    wait_async0();
#else
#pragma unroll
    for (int i = 0; i < 4; ++i) {
      int idx = tid + 256 * i;
      int r = idx >> 3, c4 = (idx & 7) << 2;
      *(v4f*)&As[r * 36 + c4] = ta[i];
    }
#pragma unroll
    for (int i = 0; i < 2; ++i) {
      int idx = tid + 256 * i;
      int r = idx >> 3, c4 = (idx & 7) << 2;
      *(v4f*)&Us[r * 36 + c4] = tu[i];
      *(v4f*)&Gs[r * 36 + c4] = tg[i];
    }
#endif
    __syncthreads();
#ifndef HAVE_ASYNC_LDS
    if (k0 + 32 < K) ldg(k0 + 32);
#endif
#pragma unroll
    for (int kk = 0; kk < 32; kk += 4) {
      v2f a[2], bU[2], bG[2];
#pragma unroll
      for (int im = 0; im < 2; ++im)
        a[im] = *(const v2f*)&As[(32 * wm + 16 * im + lm) * 36 + kk + 2 * half];
#pragma unroll
      for (int in = 0; in < 2; ++in) {
        bU[in] = *(const v2f*)&Us[(32 * wn + 16 * in + lm) * 36 + kk + 2 * half];
        bG[in] = *(const v2f*)&Gs[(32 * wn + 16 * in + lm) * 36 + kk + 2 * half];
      }
#pragma unroll
      for (int im = 0; im < 2; ++im)
#pragma unroll
        for (int in = 0; in < 2; ++in) {
          au[im][in] = wmma4(a[im], bU[in], au[im][in]);
          ag[im][in] = wmma4(a[im], bG[in], ag[im][in]);
        }
    }
  }
#pragma unroll
  for (int im = 0; im < 2; ++im)
#pragma unroll
    for (int in = 0; in < 2; ++in)
#pragma unroll
      for (int i = 0; i < 8; ++i) {
        int row = m0 + 32 * wm + 16 * im + 8 * half + i;
        int col = n0 + 32 * wn + 16 * in + lm;
        float u = au[im][in][i] + bu[col];
        float g = ag[im][in][i] + bg[col];
        Hid[(size_t)row * N + col] = siluf(g) * u;
      }
}

// =====================================================================
// LayerNorm over last dim = 128 (one wave per row)
// =====================================================================
__global__ __launch_bounds__(256) void ln128(const float* __restrict__ in,
                                             const float* __restrict__ g,
                                             const float* __restrict__ bta,
                                             float* __restrict__ out) {
  const int lane = threadIdx.x & 31, wave = threadIdx.x >> 5;
  const size_t r = (size_t)blockIdx.x * 8 + wave;
  float v[4], s = 0.f, s2 = 0.f;
#pragma unroll
  for (int j = 0; j < 4; ++j) {
    v[j] = in[r * Aa + lane + 32 * j];
    s += v[j];
    s2 += v[j] * v[j];
  }
  s = wave_sum32(s);
  s2 = wave_sum32(s2);
  float mean = s * (1.0f / Aa);
  float var = s2 * (1.0f / Aa) - mean * mean;
  float rstd = rsqrtf(var + 1e-5f);
#pragma unroll
  for (int j = 0; j < 4; ++j) {
    int c = lane + 32 * j;
    out[r * Aa + c] = (v[j] - mean) * rstd * g[c] + bta[c];
  }
}

// =====================================================================
// Flash-style pair attention:
//   aw[s,t] = silu(clip(<ain[s,:],aout[t,:]>, -5, 5)); adapt = aw @ ain
// one wave -> 16 s-rows; aw tile never touches HBM (LDS relayout only)
// =====================================================================
__global__ __launch_bounds__(256) void attn_adapt(
    const float* __restrict__ ain, const float* __restrict__ aout,
    float* __restrict__ adapt) {
  __shared__ float Ko[16 * 132];      // K tile (adapt_out rows), padded
  __shared__ float Vi[16 * 132];      // V tile (adapt_in rows)
  __shared__ float Aw[8][16 * 18];    // wave-private 16x16 aw tile, padded
  const int tid = threadIdx.x;
  const int lane = tid & 31, wave = tid >> 5;
  const int lm = lane & 15, half = lane >> 4;
  const int b = blockIdx.y;
  const float* ainb = ain + (size_t)b * Ss * Aa;
  const float* aoutb = aout + (size_t)b * Ss * Aa;
  const int s0 = blockIdx.x * 128 + wave * 16;

  v2f qa[32];  // this wave's Q (16 x 128) as WMMA A-fragments, kept in VGPRs
#pragma unroll
  for (int ks = 0; ks < 32; ++ks)
    qa[ks] = *(const v2f*)&ainb[(size_t)(s0 + lm) * Aa + 4 * ks + 2 * half];

  v8f accO[8] = {};
#pragma unroll 1
  for (int t0 = 0; t0 < Ss; t0 += 16) {
    __syncthreads();
    {  // reg-staged tile loads: issue all 4 b128 loads, then store
      const int idx0 = tid, idx1 = tid + 256;
      const int r0 = idx0 >> 5, c0 = (idx0 & 31) << 2;
      const int r1 = idx1 >> 5, c1 = (idx1 & 31) << 2;
      v4f k0 = *(const v4f*)&aoutb[(size_t)(t0 + r0) * Aa + c0];
      v4f k1 = *(const v4f*)&aoutb[(size_t)(t0 + r1) * Aa + c1];
      v4f v0 = *(const v4f*)&ainb[(size_t)(t0 + r0) * Aa + c0];
      v4f v1 = *(const v4f*)&ainb[(size_t)(t0 + r1) * Aa + c1];
      *(v4f*)&Ko[r0 * 132 + c0] = k0;
      *(v4f*)&Ko[r1 * 132 + c1] = k1;
      *(v4f*)&Vi[r0 * 132 + c0] = v0;
      *(v4f*)&Vi[r1 * 132 + c1] = v1;
    }
    __syncthreads();
    // QK^T: 16x16 tile, K-dim = 128
    v8f qk = {};
#pragma unroll
    for (int ks = 0; ks < 32; ++ks) {
      v2f kb = *(const v2f*)&Ko[lm * 132 + 4 * ks + 2 * half];
      qk = wmma4(qa[ks], kb, qk);
    }
    // activation + relayout through wave-private LDS (C-layout -> A-layout)
#pragma unroll
    for (int i = 0; i < 8; ++i) {
      float v = fminf(5.0f, fmaxf(-5.0f, qk[i]));
      Aw[wave][(8 * half + i) * 18 + lm] = siluf(v);
    }
    // PV: accumulate aw(16x16) @ Vi(16x128)
#pragma unroll
    for (int kk = 0; kk < 16; kk += 4) {
      v2f a = *(const v2f*)&Aw[wave][lm * 18 + kk + 2 * half];
#pragma unroll
      for (int n = 0; n < 8; ++n) {
        v2f bv;
        bv.x = Vi[(kk + 2 * half) * 132 + 16 * n + lm];
        bv.y = Vi[(kk + 2 * half + 1) * 132 + 16 * n + lm];
        accO[n] = wmma4(a, bv, accO[n]);
      }
    }
  }
#pragma unroll
  for (int n = 0; n < 8; ++n)
#pragma unroll
    for (int i = 0; i < 8; ++i)
      adapt[(size_t)(b * Ss + s0 + 8 * half + i) * Aa + 16 * n + lm] =
          accO[n][i];
}

// =====================================================================
// Fused expert path: mixed[t,c] = sum_e ew[t,e] * LN_c(pre[t,:] @ aW[e]^T)
// wave -> 16 token rows x full 128 cols; LN via 16-lane shfl reductions
// =====================================================================
__global__ __launch_bounds__(256) void expert_mixed(
    const float* __restrict__ pre, const float* __restrict__ aW,
    const float* __restrict__ ag, const float* __restrict__ ab,
    const float* __restrict__ ew, float* __restrict__ mixed) {
  const int tid = threadIdx.x;
  const int lane = tid & 31, wave = tid >> 5;
  const int lm = lane & 15, half = lane >> 4;
  const int t0 = blockIdx.x * 128 + wave * 16;

  v2f pa[32];
#pragma unroll
  for (int ks = 0; ks < 32; ++ks)
    pa[ks] = *(const v2f*)&pre[(size_t)(t0 + lm) * Aa + 4 * ks + 2 * half];

  v8f macc[8] = {};
#pragma unroll 1
  for (int e = 0; e < Ee; ++e) {
    v8f hacc[8] = {};
#pragma unroll
    for (int ks = 0; ks < 32; ++ks) {
#pragma unroll
      for (int n = 0; n < 8; ++n) {
        v2f bb = *(const v2f*)&aW[((size_t)e * Aa + 16 * n + lm) * Aa + 4 * ks +
                                  2 * half];
        hacc[n] = wmma4(pa[ks], bb, hacc[n]);
      }
    }
    // per-row LayerNorm over c (8 rows per lane, cols across 16 lanes x 8 n)
    float s1[8], s2[8];
#pragma unroll
    for (int i = 0; i < 8; ++i) {
      s1[i] = 0.f;
      s2[i] = 0.f;
#pragma unroll
      for (int n = 0; n < 8; ++n) {
        float v = hacc[n][i];
        s1[i] += v;
        s2[i] += v * v;
      }
      s1[i] = half_sum16(s1[i]);
      s2[i] = half_sum16(s2[i]);
    }
    float mean[8], rstd[8], wei[8];
#pragma unroll
    for (int i = 0; i < 8; ++i) {
      mean[i] = s1[i] * (1.0f / Aa);
      float var = s2[i] * (1.0f / Aa) - mean[i] * mean[i];
      rstd[i] = rsqrtf(var + 1e-5f);
      wei[i] = ew[(size_t)(t0 + 8 * half + i) * Ee + e];
    }
#pragma unroll
    for (int n = 0; n < 8; ++n) {
      float gg = ag[e * Aa + 16 * n + lm];
      float bb = ab[e * Aa + 16 * n + lm];
#pragma unroll
      for (int i = 0; i < 8; ++i) {
        float v = (hacc[n][i] - mean[i]) * rstd[i] * gg + bb;
        macc[n][i] += wei[i] * v;
      }
    }
  }
#pragma unroll
  for (int n = 0; n < 8; ++n)
#pragma unroll
    for (int i = 0; i < 8; ++i)
      mixed[(size_t)(t0 + 8 * half + i) * Aa + 16 * n + lm] = macc[n][i];
}

// =====================================================================
// Router: one wave per token; 6 length-1024 dot products; top-1 group,
// top-2 local experts; writes ew[T,8], ewsum[T], zt[T] (z-loss per token)
// =====================================================================
__global__ __launch_bounds__(256) void router(
    const float* __restrict__ x, const float* __restrict__ Wg,
    const float* __restrict__ We, float* __restrict__ ew,
    float* __restrict__ ewsum, float* __restrict__ zt) {
  const int lane = threadIdx.x & 31, wave = threadIdx.x >> 5;
  const size_t t = (size_t)blockIdx.x * 8 + wave;
  float d0 = 0.f, d1 = 0.f, l0 = 0.f, l1 = 0.f, l2 = 0.f, l3 = 0.f;
  for (int j = lane; j < Dd; j += 32) {
    float xv = x[t * Dd + j];
    d0 += xv * Wg[j];
    d1 += xv * Wg[Dd + j];
    l0 += xv * We[j];
    l1 += xv * We[Dd + j];
    l2 += xv * We[2 * Dd + j];
    l3 += xv * We[3 * Dd + j];
  }
  d0 = wave_sum32(d0);
  d1 = wave_sum32(d1);
  l0 = wave_sum32(l0);
  l1 = wave_sum32(l1);
  l2 = wave_sum32(l2);
  l3 = wave_sum32(l3);
  if (lane == 0) {
    // group softmax (2) + top-1 (ties -> lower index)
    float gm = fmaxf(d0, d1);
    float p0 = expf(d0 - gm), p1 = expf(d1 - gm);
    float ps = p0 + p1;
    int g = (p0 >= p1) ? 0 : 1;
    float gw = ((p0 >= p1) ? p0 : p1) / ps;
    // local softmax (4) + top-2
    float l[4] = {l0, l1, l2, l3};
    float lmx = fmaxf(fmaxf(l[0], l[1]), fmaxf(l[2], l[3]));
    float p[4], sum = 0.f;
#pragma unroll
    for (int i = 0; i < 4; ++i) {
      p[i] = expf(l[i] - lmx);
      sum += p[i];
    }
#pragma unroll
    for (int i = 0; i < 4; ++i) p[i] /= sum;
    int i1 = 0;
#pragma unroll
    for (int i = 1; i < 4; ++i)
      if (p[i] > p[i1]) i1 = i;
    int i2 = -1;
#pragma unroll
    for (int i = 0; i < 4; ++i)
      if (i != i1 && (i2 < 0 || p[i] > p[i2])) i2 = i;
    float tsum = p[i1] + p[i2] + 1e-7f;
    float w1 = gw * p[i1] / tsum, w2 = gw * p[i2] / tsum;
    float row[8] = {0.f, 0.f, 0.f, 0.f, 0.f, 0.f, 0.f, 0.f};
    row[g * 4 + i1] = w1;
    row[g * 4 + i2] = w2;
#pragma unroll
    for (int e = 0; e < 8; ++e) ew[t * Ee + e] = row[e];
    ewsum[t] = w1 + w2;
    zt[t] = 0.5f * (d0 * d0 + d1 * d1) +
            0.25f * (l0 * l0 + l1 * l1 + l2 * l2 + l3 * l3);
  }
}

// deterministic loss reduction (fixed tree order; no float atomics)
__global__ __launch_bounds__(256) void finalize_loss(
    const float* __restrict__ ew, const float* __restrict__ zt,
    float* __restrict__ loss_out) {
  __shared__ float red[256];
  const int tid = threadIdx.x;
  float z = 0.f, le[8] = {0.f, 0.f, 0.f, 0.f, 0.f, 0.f, 0.f, 0.f};
  for (int t = tid; t < Tt; t += 256) {
    z += zt[t];
#pragma unroll
    for (int e = 0; e < 8; ++e) le[e] += ew[(size_t)t * Ee + e];
  }
  red[tid] = z;
  __syncthreads();
  for (int s = 128; s > 0; s >>= 1) {
    if (tid < s) red[tid] += red[tid + s];
    __syncthreads();
  }
  float zsum = red[0];
  __syncthreads();
  float loads[8];
  for (int e = 0; e < 8; ++e) {
    red[tid] = le[e];
    __syncthreads();
    for (int s = 128; s > 0; s >>= 1) {
      if (tid < s) red[tid] += red[tid + s];
      __syncthreads();
    }
    loads[e] = red[0];
    __syncthreads();
  }
  if (tid == 0) {
    float s = 0.f;
#pragma unroll
    for (int e = 0; e < 8; ++e) s += loads[e];
    float tgt = s / 8.0f;
    float lb = 0.f;
#pragma unroll
    for (int e = 0; e < 8; ++e) {
      float d = loads[e] - tgt;
      lb += d * d;
    }
    lb /= 8.0f;
    loss_out[0] = 0.001f * (lb + zsum / (float)Tt);
  }
}

// W_eprojT[a,h] = W_eproj[h,a]
__global__ __launch_bounds__(256) void transp_HA(const float* __restrict__ in,
                                                 float* __restrict__ out) {
  int idx = blockIdx.x * 256 + threadIdx.x;
  if (idx < Hh * Aa) {
    int h = idx / Aa, a = idx % Aa;
    out[(size_t)a * Hh + h] = in[idx];
  }
}

// =====================================================================
extern "C" void kernel_launch(void* const* d_in, const int* in_sizes, int n_in,
                              void* d_out, int out_size, void* d_ws,
                              size_t ws_size, hipStream_t stream) {
  const float* x = (const float*)d_in[0];
  const float* Wg = (const float*)d_in[1];
  const float* We = (const float*)d_in[2];
  const float* W_up = (const float*)d_in[3];
  const float* b_up = (const float*)d_in[4];
  const float* W_gate = (const float*)d_in[5];
  const float* b_gate = (const float*)d_in[6];
  const float* W_down = (const float*)d_in[7];
  const float* b_down = (const float*)d_in[8];
  const float* W_pre = (const float*)d_in[9];
  const float* b_pre = (const float*)d_in[10];
  const float* W_post = (const float*)d_in[11];
  const float* b_post = (const float*)d_in[12];
  const float* ln_g = (const float*)d_in[13];
  const float* ln_b = (const float*)d_in[14];
  const float* W_aproj = (const float*)d_in[15];
  const float* adapter_W = (const float*)d_in[16];
  const float* adapter_g = (const float*)d_in[17];
  const float* adapter_b = (const float*)d_in[18];
  const float* W_eproj = (const float*)d_in[19];
  const float* W_oproj = (const float*)d_in[20];
  float* out = (float*)d_out;  // [T*D] then [1] loss

  // ---- workspace layout (floats) ----
  float* p = (float*)d_ws;
  float* hidden = p;       p += (size_t)Tt * Hh;   // 64 MB
  float* pre = p;          p += (size_t)Tt * Aa;
  float* adapt_in = p;     p += (size_t)Tt * Aa;
  float* post = p;         p += (size_t)Tt * Aa;
  float* adapt_out = p;    p += (size_t)Tt * Aa;
  float* adapt = p;        p += (size_t)Tt * Aa;
  float* mixed = p;        p += (size_t)Tt * Aa;
  float* ew = p;           p += (size_t)Tt * Ee;
  float* ewsum = p;        p += (size_t)Tt;
  float* zt = p;           p += (size_t)Tt;
  float* WeT = p;          p += (size_t)Aa * Hh;
  float* Wcomb = p;        p += (size_t)Dd * Aa;

  // 1) router (ew, ewsum, per-token z)
  router<<<Tt / 8, 256, 0, stream>>>(x, Wg, We, ew, ewsum, zt);
  // 2) pre = x @ W_pre^T + b_pre ; adapt_in = LN(pre)
  gemm_xwt<<<dim3(Aa / 64, Tt / 128), 256, 0, stream>>>(
      x, W_pre, b_pre, pre, nullptr, Tt, Aa, Dd, 0.f, EP_STORE);
  ln128<<<Tt / 8, 256, 0, stream>>>(pre, ln_g, ln_b, adapt_in);
  // 3) hidden = silu(x@W_gate^T+bg) * (x@W_up^T+bu)
  gemm_upgate<<<dim3(Hh / 64, Tt / 128), 256, 0, stream>>>(
      x, W_up, b_up, W_gate, b_gate, hidden, Tt, Hh, Dd);
  // 4) post = hidden @ W_post^T + b_post ; adapt_out = LN(post)
  gemm_xwt<<<dim3(Aa / 64, Tt / 128), 256, 0, stream>>>(
      hidden, W_post, b_post, post, nullptr, Tt, Aa, Hh, 0.f, EP_STORE);
  ln128<<<Tt / 8, 256, 0, stream>>>(post, ln_g, ln_b, adapt_out);
  // 5) adapt = silu(clip(ain@aout^T)) @ ain   (flash-style, per batch)
  attn_adapt<<<dim3(Ss / 128, Bb), 256, 0, stream>>>(adapt_in, adapt_out,
                                                     adapt);
  // 6) hidden += 0.1 * adapt @ W_aproj^T
  gemm_xwt<<<dim3(Hh / 64, Tt / 128), 256, 0, stream>>>(
      adapt, W_aproj, nullptr, hidden, nullptr, Tt, Hh, Aa, 0.1f, EP_ADDSC);
  // 7) shared = hidden @ W_down^T + b_down  -> written straight into d_out
  gemm_xwt<<<dim3(Dd / 64, Tt / 128), 256, 0, stream>>>(
      hidden, W_down, b_down, out, nullptr, Tt, Dd, Hh, 0.f, EP_STORE);
  // 8) W_comb = W_oproj @ W_eproj  (via transpose so GEMM stays A@W^T)
  transp_HA<<<(Hh * Aa) / 256, 256, 0, stream>>>(W_eproj, WeT);
  gemm_xwt<<<dim3(Aa / 64, Dd / 128), 256, 0, stream>>>(
      W_oproj, WeT, nullptr, Wcomb, nullptr, Dd, Aa, Hh, 0.f, EP_STORE);
  // 9) mixed[t,c] = sum_e ew[t,e] * LN(pre[t]@adapter_W[e]^T)
  expert_mixed<<<Tt / 128, 256, 0, stream>>>(pre, adapter_W, adapter_g,
                                             adapter_b, ew, mixed);
  // 10) out = out*ewsum[t] + 0.1 * mixed @ W_comb^T   (in place)
  gemm_xwt<<<dim3(Dd / 64, Tt / 128), 256, 0, stream>>>(
      mixed, Wcomb, nullptr, out, ewsum, Tt, Dd, Aa, 0.1f, EP_FINAL);
  // 11) router_loss -> out[T*D]
  finalize_loss<<<1, 256, 0, stream>>>(ew, zt, out + (size_t)Tt * Dd);
}